// MyConv2D_40913858462313
// MI455X (gfx1250) — compile-verified
//
#include <hip/hip_runtime.h>

// Problem constants (from the reference)
#define BATCH 32
#define CIN   64
#define COUT  128
#define HH    128
#define WW    128
#define OH    126
#define OW    126
#define PP    (OH * OW)        // 15876 output positions
#define KDIM  (CIN * 9)        // 576 = GEMM K dimension
#define KSTEPS (KDIM / 4)      // 144 WMMA K-steps of 4
#define NTILES ((PP + 15) / 16) // 993 N-tiles of 16

typedef __attribute__((ext_vector_type(2))) float v2f;
typedef __attribute__((ext_vector_type(4))) float v4f;
typedef __attribute__((ext_vector_type(8))) float v8f;

// ---------------------------------------------------------------------------
// Pass 1: xs[c,h,w] = sum_b x[b,c,h,w].  134 MB read / 4 MB write, float4.
// x is streamed exactly once -> non-temporal loads so it doesn't evict the
// reused buffers from L2. xs is re-read by the conv kernel -> regular store.
// 262144 threads, each owns one float4 of xs.
// ---------------------------------------------------------------------------
__global__ void __launch_bounds__(256) batch_sum_kernel(
    const float* __restrict__ x, float* __restrict__ xs) {
    long t = (long)blockIdx.x * blockDim.x + threadIdx.x;
    long idx = t * 4;
    v4f acc = {0.f, 0.f, 0.f, 0.f};
    #pragma unroll
    for (int b = 0; b < BATCH; ++b) {
        acc += __builtin_nontemporal_load(
            (const v4f*)(x + (long)b * (CIN * HH * WW) + idx));
    }
    *(v4f*)(xs + idx) = acc;
}

// ---------------------------------------------------------------------------
// Pass 2: swizzle weight [COUT, KDIM] into per-wave A-fragment layout for
// V_WMMA_F32_16X16X4_F32 (ISA 7.12.2, 32-bit A 16x4):
//   lane l holds row m = l%16; VGPR0 = K = 2*(l/16), VGPR1 = K+1.
// Stored as wswz[((mt*KSTEPS + ks)*32 + lane)] : v2f  => coalesced loads.
// 36864 threads, one v2f each.
// ---------------------------------------------------------------------------
__global__ void __launch_bounds__(256) wswz_kernel(
    const float* __restrict__ wt, float* __restrict__ wswz) {
    int t    = blockIdx.x * blockDim.x + threadIdx.x;   // [0, 8*144*32)
    int lane = t & 31;
    int ks   = (t >> 5) % KSTEPS;
    int mt   = t / (32 * KSTEPS);
    int m    = mt * 16 + (lane & 15);
    int k    = ks * 4 + ((lane >> 4) << 1);
    v2f v = { wt[m * KDIM + k], wt[m * KDIM + k + 1] };
    *(v2f*)(wswz + (size_t)t * 2) = v;
}

// ---------------------------------------------------------------------------
// Pass 3: y_base[co, p] = bias[co] + sum_k Wmat[co,k] * im2col(xs)[k,p]
// One wave per 16-wide N-tile; 8 accumulators cover all COUT=128 rows.
// B fragment (4x16, K x N): VGPR0 = Im[k0][n], VGPR1 = Im[k0+1][n],
//   k0 = ks*4 + 2*(lane/16), n = lane%16  (mirrors the A layout).
// C/D (16x16): lane l, VGPR r -> co = 16*mt + 8*(l/16) + r, n = l%16.
// ---------------------------------------------------------------------------
__global__ void __launch_bounds__(128) conv_wmma_kernel(
    const float* __restrict__ xs, const float* __restrict__ wswz,
    const float* __restrict__ bias, float* __restrict__ ybase) {
    int wave = (int)((blockIdx.x * blockDim.x + threadIdx.x) >> 5);
    int lane = threadIdx.x & 31;
    if (wave >= NTILES) return;                 // wave-uniform: EXEC stays full

    int nloc  = lane & 15;
    int khalf = lane >> 4;                      // 0: K={0,1}, 1: K={2,3}
    int p  = wave * 16 + nloc;
    int pc = p < PP ? p : PP - 1;               // clamp loads for tail tile
    int oh = pc / OW;
    int ow = pc - oh * OW;
    const float* xbase = xs + oh * WW + ow;     // + ci*HH*WW + kh*WW + kw
    const v2f*   afrag = (const v2f*)wswz;

    v8f acc[8];
    #pragma unroll
    for (int mt = 0; mt < 8; ++mt) acc[mt] = (v8f){0.f,0.f,0.f,0.f,0.f,0.f,0.f,0.f};

    for (int ks = 0; ks < KSTEPS; ++ks) {
        // Gather B fragment: decode k -> (ci, kh, kw)
        int k0  = ks * 4 + khalf * 2;
        int k1  = k0 + 1;
        int ci0 = k0 / 9; int r0 = k0 - ci0 * 9; int kh0 = r0 / 3; int kw0 = r0 - kh0 * 3;
        int ci1 = k1 / 9; int r1 = k1 - ci1 * 9; int kh1 = r1 / 3; int kw1 = r1 - kh1 * 3;
        v2f bf = { xbase[ci0 * (HH * WW) + kh0 * WW + kw0],
                   xbase[ci1 * (HH * WW) + kh1 * WW + kw1] };

        const v2f* arow = afrag + (size_t)ks * 32 + lane;
        #pragma unroll
        for (int mt = 0; mt < 8; ++mt) {
            v2f af = arow[(size_t)mt * KSTEPS * 32];
            acc[mt] = __builtin_amdgcn_wmma_f32_16x16x4_f32(
                false, af, false, bf, (short)0, acc[mt], false, false);
        }
    }

    if (p < PP) {                               // predicate only the stores
        #pragma unroll
        for (int mt = 0; mt < 8; ++mt) {
            int co0 = mt * 16 + khalf * 8;
            #pragma unroll
            for (int r = 0; r < 8; ++r) {
                int co = co0 + r;
                ybase[(size_t)co * PP + p] = acc[mt][r] + bias[co];
            }
        }
    }
}

// ---------------------------------------------------------------------------
// Pass 4: out[b, co, p] = ybase[co, p]. 260 MB of streaming stores — the
// dominant cost — fully coalesced float4 with non-temporal hint (write-once,
// never re-read); ybase (8.1 MB) stays resident in the 192 MB L2.
// 16,257,024 threads, one float4 each (PP divisible by 4).
// ---------------------------------------------------------------------------
__global__ void __launch_bounds__(256) bcast_kernel(
    const float* __restrict__ ybase, float* __restrict__ out) {
    size_t t = (size_t)blockIdx.x * blockDim.x + threadIdx.x;
    size_t q = t % (size_t)(COUT * PP / 4);
    v4f v = *(const v4f*)(ybase + q * 4);
    __builtin_nontemporal_store(v, (v4f*)(out + t * 4));
}

extern "C" void kernel_launch(void* const* d_in, const int* in_sizes, int n_in,
                              void* d_out, int out_size, void* d_ws, size_t ws_size,
                              hipStream_t stream) {
    const float* x    = (const float*)d_in[0];   // [32, 64, 128, 128]
    const float* wt   = (const float*)d_in[1];   // [128, 64, 3, 3]
    const float* bias = (const float*)d_in[2];   // [128, 1, 1]
    float*       out  = (float*)d_out;           // [32, 128, 126, 126]

    float* xs    = (float*)d_ws;                 // 1,048,576 floats (4 MB)
    float* wswz  = xs + (size_t)CIN * HH * WW;   //    73,728 floats (288 KB)
    float* ybase = wswz + 8 * KSTEPS * 32 * 2;   // 2,032,128 floats (8.1 MB)

    batch_sum_kernel<<<1024, 256, 0, stream>>>(x, xs);
    wswz_kernel<<<144, 256, 0, stream>>>(wt, wswz);
    conv_wmma_kernel<<<(NTILES + 3) / 4, 128, 0, stream>>>(xs, wswz, bias, ybase);
    bcast_kernel<<<(COUT * PP * BATCH / 4) / 256, 256, 0, stream>>>(ybase, out);
}